// Generator_59966333387169
// MI455X (gfx1250) — compile-verified
//
#include <hip/hip_runtime.h>
#include <math.h>

typedef float v2f __attribute__((ext_vector_type(2)));
typedef float v8f __attribute__((ext_vector_type(8)));

#define LDS_FENCE() asm volatile("s_wait_dscnt 0" ::: "memory")

constexpr int B_TOT = 131072;
constexpr int S_LEN = 30;
constexpr int D_IN  = 4;
constexpr int V1 = 21;
constexpr int V2 = 11;

constexpr int WAVES = 8;          // 256 threads / block
constexpr int WBR   = 320;        // weight floats per branch in LDS
constexpr int W_TOT = 2 * WBR;    // 640
constexpr int PW    = 3072;       // per-wave scratch floats
// per-wave scratch offsets (floats)
constexpr int OQ = 0, OKK = 512, OV = 1024, OA = 2048;
// weight offsets within a branch (floats)
constexpr int WQo = 0, WKo = 64, WVo = 128, BQo = 256, BKo = 272, BVo = 288;

__device__ __forceinline__ v8f wmma4(v2f a, v2f b, v8f c) {
  // V_WMMA_F32_16X16X4_F32 : D = A(16x4) * B(4x16) + C(16x16)
  return __builtin_amdgcn_wmma_f32_16x16x4_f32(false, a, false, b, (short)0, c,
                                               false, false);
}

__device__ __forceinline__ float fast_rcp(float x) {
#if __has_builtin(__builtin_amdgcn_rcpf)
  return __builtin_amdgcn_rcpf(x);
#else
  return 1.f / x;
#endif
}

__device__ __forceinline__ float fast_tanh(float x) {
#if __has_builtin(__builtin_amdgcn_tanhf)
  return __builtin_amdgcn_tanhf(x);
#else
  // tanh(x) = sign(x) * (1 - 2*e / (1 + e)),  e = exp(-2|x|)  (overflow-safe)
  float ax = fabsf(x);
  float e  = __expf(-2.f * ax);
  float r  = 1.f - 2.f * e * fast_rcp(1.f + e);
  return copysignf(r, x);
#endif
}

__global__ void __launch_bounds__(256)
attn_dual_kernel(const float* __restrict__ x,
                 const float* __restrict__ q1w, const float* __restrict__ q1b,
                 const float* __restrict__ k1w, const float* __restrict__ k1b,
                 const float* __restrict__ v1w, const float* __restrict__ v1b,
                 const float* __restrict__ q2w, const float* __restrict__ q2b,
                 const float* __restrict__ k2w, const float* __restrict__ k2b,
                 const float* __restrict__ v2w, const float* __restrict__ v2b,
                 float* __restrict__ out1, float* __restrict__ out2) {
  extern __shared__ float smem[];

  // ---- one-time: stage zero-padded weights into LDS ----
  const float* wsrc[12] = {q1w, q1b, k1w, k1b, v1w, v1b,
                           q2w, q2b, k2w, k2b, v2w, v2b};
  for (int i = threadIdx.x; i < W_TOT; i += blockDim.x) {
    int br = i / WBR, r = i % WBR;
    const float* qw = wsrc[br * 6 + 0];
    const float* qb = wsrc[br * 6 + 1];
    const float* kw = wsrc[br * 6 + 2];
    const float* kb = wsrc[br * 6 + 3];
    const float* vw = wsrc[br * 6 + 4];
    const float* vb = wsrc[br * 6 + 5];
    int V = br ? V2 : V1;
    float val;
    if (r < 64)       val = qw[r];                       // [4][16]
    else if (r < 128) val = kw[r - 64];                  // [4][16]
    else if (r < 256) {                                  // vw padded [4][32]
      int rr = (r - 128) >> 5, cc = (r - 128) & 31;
      val = (cc < V) ? vw[rr * V + cc] : 0.f;
    } else if (r < 272) val = qb[r - 256];
    else if (r < 288)   val = kb[r - 272];
    else { int cc = r - 288; val = (cc < V) ? vb[cc] : 0.f; }  // vb padded [32]
    smem[i] = val;
  }
  __syncthreads();

  const int lane = threadIdx.x & 31;
  const int wave = threadIdx.x >> 5;
  const int half = lane >> 4;   // 0 or 1 (which 16-lane half)
  const int l16  = lane & 15;

  float* lw = smem + W_TOT + wave * PW;
  float* lq = lw + OQ;    // [32][16] q rows
  float* lk = lw + OKK;   // [32][16] k rows
  float* lv = lw + OV;    // [32][32] v rows (cols zero-padded)
  float* la = lw + OA;    // [32][32] attn (pad cols = 0)

  const int gw = blockIdx.x * WAVES + wave;
  const int nw = gridDim.x * WAVES;

  for (int b = gw; b < B_TOT; b += nw) {
    // x as WMMA A-operand (16x4): lane m holds row m; K split across halves.
    const float* xb = x + (size_t)b * (S_LEN * D_IN);
    v2f xa[2];
#pragma unroll
    for (int mt = 0; mt < 2; ++mt) {
      int s = l16 + 16 * mt;
      if (s > S_LEN - 1) s = S_LEN - 1;          // clamp pad rows (finite)
      xa[mt] = *(const v2f*)(xb + s * 4 + 2 * half);
    }

#pragma unroll
    for (int br = 0; br < 2; ++br) {
      const float* wb = smem + br * WBR;
      const int V  = br ? V2 : V1;
      const int NT = br ? 1 : 2;                 // N tiles for v/out
      float* op = br ? out2 : out1;

      LDS_FENCE();  // WAR vs previous branch's lv/la reads

      // ---- projections: q,k,v = x @ W + bias (WMMA, K=4 in one step) ----
#pragma unroll
      for (int mt = 0; mt < 2; ++mt) {
        { // q
          float bqv = wb[BQo + l16];
          v8f c;
#pragma unroll
          for (int r = 0; r < 8; ++r) c[r] = bqv;
          v2f bm;
          bm.x = wb[WQo + (2 * half + 0) * 16 + l16];
          bm.y = wb[WQo + (2 * half + 1) * 16 + l16];
          c = wmma4(xa[mt], bm, c);
#pragma unroll
          for (int r = 0; r < 8; ++r)
            lq[(16 * mt + r + 8 * half) * 16 + l16] = c[r];
        }
        { // k
          float bkv = wb[BKo + l16];
          v8f c;
#pragma unroll
          for (int r = 0; r < 8; ++r) c[r] = bkv;
          v2f bm;
          bm.x = wb[WKo + (2 * half + 0) * 16 + l16];
          bm.y = wb[WKo + (2 * half + 1) * 16 + l16];
          c = wmma4(xa[mt], bm, c);
#pragma unroll
          for (int r = 0; r < 8; ++r)
            lk[(16 * mt + r + 8 * half) * 16 + l16] = c[r];
        }
        for (int nt = 0; nt < NT; ++nt) { // v (zero-padded columns)
          float bvv = wb[BVo + 16 * nt + l16];
          v8f c;
#pragma unroll
          for (int r = 0; r < 8; ++r) c[r] = bvv;
          v2f bm;
          bm.x = wb[WVo + (2 * half + 0) * 32 + 16 * nt + l16];
          bm.y = wb[WVo + (2 * half + 1) * 32 + 16 * nt + l16];
          c = wmma4(xa[mt], bm, c);
#pragma unroll
          for (int r = 0; r < 8; ++r)
            lv[(16 * mt + r + 8 * half) * 32 + 16 * nt + l16] = c[r];
        }
      }
      LDS_FENCE();

      // ---- scores = q @ k^T  (2x2 tiles of 16x16, K=16 over 4 steps) ----
      v8f sc[2][2];
#pragma unroll
      for (int mt = 0; mt < 2; ++mt) {
        v2f am[4];
#pragma unroll
        for (int c4 = 0; c4 < 4; ++c4)
          am[c4] = *(const v2f*)(lq + (l16 + 16 * mt) * 16 + 4 * c4 + 2 * half);
#pragma unroll
        for (int nt = 0; nt < 2; ++nt) {
          v8f acc = {0.f, 0.f, 0.f, 0.f, 0.f, 0.f, 0.f, 0.f};
#pragma unroll
          for (int c4 = 0; c4 < 4; ++c4) {
            v2f bm = *(const v2f*)(lk + (l16 + 16 * nt) * 16 + 4 * c4 + 2 * half);
            acc = wmma4(am[c4], bm, acc);
          }
          sc[mt][nt] = acc;
        }
      }

      // ---- softmax over t (row spans 16 lanes of one half) ----
      const float NEG_INF = -__builtin_inff();
#pragma unroll
      for (int mt = 0; mt < 2; ++mt) {
#pragma unroll
        for (int r = 0; r < 8; ++r) {
          float s0 = sc[mt][0][r];
          float s1 = (l16 < 14) ? sc[mt][1][r] : NEG_INF;  // mask t=30,31
          float m = fmaxf(s0, s1);
          m = fmaxf(m, __shfl_xor(m, 1, 32));
          m = fmaxf(m, __shfl_xor(m, 2, 32));
          m = fmaxf(m, __shfl_xor(m, 4, 32));
          m = fmaxf(m, __shfl_xor(m, 8, 32));
          float e0 = __expf(s0 - m);
          float e1 = (l16 < 14) ? __expf(s1 - m) : 0.f;
          float sum = e0 + e1;
          sum += __shfl_xor(sum, 1, 32);
          sum += __shfl_xor(sum, 2, 32);
          sum += __shfl_xor(sum, 4, 32);
          sum += __shfl_xor(sum, 8, 32);
          float inv = fast_rcp(sum);   // sum >= 1 always (max-subtracted)
          int row = 16 * mt + r + 8 * half;
          la[row * 32 + l16]      = e0 * inv;
          la[row * 32 + 16 + l16] = e1 * inv;   // zero for pad cols
        }
      }
      LDS_FENCE();

      // ---- out = attn @ v  (K=32 padded, pad attn cols are 0) ----
#pragma unroll
      for (int mt = 0; mt < 2; ++mt) {
        v2f am[8];
#pragma unroll
        for (int c = 0; c < 8; ++c)
          am[c] = *(const v2f*)(la + (l16 + 16 * mt) * 32 + 4 * c + 2 * half);
        for (int nt = 0; nt < NT; ++nt) {
          v8f acc = {0.f, 0.f, 0.f, 0.f, 0.f, 0.f, 0.f, 0.f};
#pragma unroll
          for (int c = 0; c < 8; ++c) {
            int kr = 4 * c + 2 * half;
            v2f bm;
            bm.x = lv[(kr + 0) * 32 + 16 * nt + l16];
            bm.y = lv[(kr + 1) * 32 + 16 * nt + l16];
            acc = wmma4(am[c], bm, acc);
          }
#pragma unroll
          for (int r = 0; r < 8; ++r) {
            int row = 16 * mt + r + 8 * half;
            int col = 16 * nt + l16;
            if (row < S_LEN && col < V)
              op[(size_t)b * (S_LEN * V) + row * V + col] = fast_tanh(acc[r]);
          }
        }
      }
    }
  }
}

extern "C" void kernel_launch(void* const* d_in, const int* in_sizes, int n_in,
                              void* d_out, int out_size, void* d_ws, size_t ws_size,
                              hipStream_t stream) {
  const float* x   = (const float*)d_in[0];
  const float* q1w = (const float*)d_in[1];
  const float* q1b = (const float*)d_in[2];
  const float* k1w = (const float*)d_in[3];
  const float* k1b = (const float*)d_in[4];
  const float* v1w = (const float*)d_in[5];
  const float* v1b = (const float*)d_in[6];
  const float* q2w = (const float*)d_in[7];
  const float* q2b = (const float*)d_in[8];
  const float* k2w = (const float*)d_in[9];
  const float* k2b = (const float*)d_in[10];
  const float* v2w = (const float*)d_in[11];
  const float* v2b = (const float*)d_in[12];

  float* out1 = (float*)d_out;
  float* out2 = out1 + (size_t)B_TOT * S_LEN * V1;

  const int blocks = 2048;
  const int threads = WAVES * 32;
  const size_t shmem = (size_t)(W_TOT + WAVES * PW) * sizeof(float);
  attn_dual_kernel<<<blocks, threads, shmem, stream>>>(
      x, q1w, q1b, k1w, k1b, v1w, v1b,
      q2w, q2b, k2w, k2b, v2w, v2b, out1, out2);
}